// PWLUBase_90486370992223
// MI455X (gfx1250) — compile-verified
//
#include <hip/hip_runtime.h>

// PWLU forward, channelwise. x: [64,128,56,56] f32, points: [128,7] f32.
// Region-wise linear refactor: out = A[r]*x + B[r],
//   r = floor(clamp(x*(3/BOUND) + 3, 0, 0.999*6)),
//   A[r] = (p[r+1]-p[r]) * 3/BOUND,  B[r] = p[r] + (3-r)*(p[r+1]-p[r]).
// One (batch,channel) plane of 56*56=3136 contiguous floats per block; channel
// is block-uniform so the 6 (A,B) pairs live in LDS. 128-bit non-temporal
// loads/stores: 205MB total stream > 192MB L2, touched exactly once.

typedef float v4f __attribute__((ext_vector_type(4)));
typedef float v2f __attribute__((ext_vector_type(2)));

#define PWLU_BOUND 2.5f
#define NPTS       7
#define NREG       6
#define PLANE      3136        // 56*56
#define PLANE_V4   (PLANE/4)   // 784 = 3*256 + 16
#define BLOCK      256         // 8 wave32 waves

__global__ __launch_bounds__(BLOCK) void pwlu_fwd_kernel(
    const float* __restrict__ x,
    const float* __restrict__ points,
    float* __restrict__ out,
    int cmask)                  // nchannels - 1 (nchannels is a power of two)
{
    __shared__ v2f ab[NREG];    // (A, B) per region for this block's channel

    const int plane = blockIdx.x;          // flat (b, c) plane index
    const int c     = plane & cmask;       // channel: single s_and, no idiv
    const int tid   = threadIdx.x;

    if (tid < NREG) {
        float pl = points[c * NPTS + tid];
        float pr = points[c * NPTS + tid + 1];
        float dp = pr - pl;
        v2f v;
        v.x = dp * (3.0f / PWLU_BOUND);                     // A
        v.y = __builtin_fmaf(3.0f - (float)tid, dp, pl);    // B
        ab[tid] = v;
    }
    __syncthreads();

    const v4f* __restrict__ xin = (const v4f*)(x + (size_t)plane * PLANE);
    v4f*       __restrict__ o   = (v4f*)(out + (size_t)plane * PLANE);

    const float scale = 3.0f / PWLU_BOUND;     // 1.2
    const float tmax  = 0.999f * (float)NREG;  // 5.994

    // Issue all three streaming loads up front (one clause, 3 in flight).
    v4f v0 = __builtin_nontemporal_load(&xin[tid]);
    v4f v1 = __builtin_nontemporal_load(&xin[tid + BLOCK]);
    v4f v2 = __builtin_nontemporal_load(&xin[tid + 2 * BLOCK]);

    v4f r0, r1, r2;
    #pragma unroll
    for (int k = 0; k < 4; ++k) {
        {
            float xv = v0[k];
            float t  = fminf(fmaxf(__builtin_fmaf(xv, scale, 3.0f), 0.0f), tmax);
            v2f  p   = ab[(int)t];
            r0[k]    = __builtin_fmaf(p.x, xv, p.y);
        }
        {
            float xv = v1[k];
            float t  = fminf(fmaxf(__builtin_fmaf(xv, scale, 3.0f), 0.0f), tmax);
            v2f  p   = ab[(int)t];
            r1[k]    = __builtin_fmaf(p.x, xv, p.y);
        }
        {
            float xv = v2[k];
            float t  = fminf(fmaxf(__builtin_fmaf(xv, scale, 3.0f), 0.0f), tmax);
            v2f  p   = ab[(int)t];
            r2[k]    = __builtin_fmaf(p.x, xv, p.y);
        }
    }
    __builtin_nontemporal_store(r0, &o[tid]);
    __builtin_nontemporal_store(r1, &o[tid + BLOCK]);
    __builtin_nontemporal_store(r2, &o[tid + 2 * BLOCK]);

    // Tail: 784 - 3*256 = 16 v4 elements.
    if (tid < PLANE_V4 - 3 * BLOCK) {
        v4f v = __builtin_nontemporal_load(&xin[tid + 3 * BLOCK]);
        v4f r;
        #pragma unroll
        for (int k = 0; k < 4; ++k) {
            float xv = v[k];
            float t  = fminf(fmaxf(__builtin_fmaf(xv, scale, 3.0f), 0.0f), tmax);
            v2f  p   = ab[(int)t];
            r[k]     = __builtin_fmaf(p.x, xv, p.y);
        }
        __builtin_nontemporal_store(r, &o[tid + 3 * BLOCK]);
    }
}

extern "C" void kernel_launch(void* const* d_in, const int* in_sizes, int n_in,
                              void* d_out, int out_size, void* d_ws, size_t ws_size,
                              hipStream_t stream) {
    const float* x      = (const float*)d_in[0];
    const float* points = (const float*)d_in[1];
    float*       out    = (float*)d_out;

    const int nchannels = in_sizes[1] / NPTS;   // 128 (power of two)
    const int nplanes   = in_sizes[0] / PLANE;  // 64*128 = 8192

    pwlu_fwd_kernel<<<nplanes, BLOCK, 0, stream>>>(x, points, out, nchannels - 1);
}